// GALA_45509473469002
// MI455X (gfx1250) — compile-verified
//
#include <hip/hip_runtime.h>

typedef __attribute__((ext_vector_type(2))) float v2f;
typedef __attribute__((ext_vector_type(8))) float v8f;

// ---------------------------------------------------------------------------
// Prologue: degree + symmetric-normalization coefficients
// ---------------------------------------------------------------------------
__global__ void k_zero_i32(int* __restrict__ p, int n) {
    int i = blockIdx.x * blockDim.x + threadIdx.x;
    if (i < n) p[i] = 0;
}

__global__ void k_count_indeg(const int* __restrict__ col, int* __restrict__ indeg, int e) {
    int i = blockIdx.x * blockDim.x + threadIdx.x;
    if (i < e) atomicAdd(&indeg[col[i]], 1);
}

// smooth: deg = indeg + 1 (A+I);  sharp: deg = indeg + 2 (A+2I, D from weights)
__global__ void k_dinv(const int* __restrict__ indeg,
                       float* __restrict__ dinv_s, float* __restrict__ dinv_h, int n) {
    int i = blockIdx.x * blockDim.x + threadIdx.x;
    if (i < n) {
        float d = (float)indeg[i];
        dinv_s[i] = rsqrtf(d + 1.0f);
        dinv_h[i] = rsqrtf(d + 2.0f);
    }
}

// ---------------------------------------------------------------------------
// WMMA fp32 GEMM:  H[N x DOUT] = act(X[N x DIN]) @ W[DIN x DOUT]
// One wave computes a 16-row x DOUT strip; NT = DOUT/16 accumulators.
// Uses V_WMMA_F32_16X16X4_F32 (exact fp32 matrix op).
// ---------------------------------------------------------------------------
template <int DIN, int NT, bool RELU_IN>
__global__ void k_gemm_wmma(const float* __restrict__ X, const float* __restrict__ W,
                            float* __restrict__ H, int N) {
    constexpr int DOUT = NT * 16;
    const int lane = threadIdx.x & 31;
    const int wave = threadIdx.x >> 5;
    const int tile = blockIdx.x * (blockDim.x >> 5) + wave;
    const int rtiles = (N + 15) >> 4;
    if (tile >= rtiles) return;                    // wave-uniform exit (EXEC stays all-1 below)

    const int mrow  = tile * 16 + (lane & 15);     // A-row this lane feeds
    const int mload = mrow < N ? mrow : N - 1;     // clamp (N%16==0 in practice)
    const int kb    = (lane >> 4) * 2;             // lanes 0-15: K0/K1, lanes 16-31: K2/K3
    const int nloc  = lane & 15;                   // B/C/D column within 16-wide tile

    v8f acc[NT];
#pragma unroll
    for (int j = 0; j < NT; ++j) {
        v8f z = {0.f, 0.f, 0.f, 0.f, 0.f, 0.f, 0.f, 0.f};
        acc[j] = z;
    }

    const float* xr = X + (size_t)mload * DIN;
#pragma unroll 4
    for (int k = 0; k < DIN; k += 4) {
        float a0 = xr[k + kb];
        float a1 = xr[k + kb + 1];
        if (RELU_IN) { a0 = fmaxf(a0, 0.0f); a1 = fmaxf(a1, 0.0f); }
        v2f A = {a0, a1};
#pragma unroll
        for (int j = 0; j < NT; ++j) {
            const int n0 = j * 16 + nloc;
            v2f B = {W[(size_t)(k + kb) * DOUT + n0],
                     W[(size_t)(k + kb + 1) * DOUT + n0]};
            acc[j] = __builtin_amdgcn_wmma_f32_16x16x4_f32(
                false, A, false, B, (short)0, acc[j], false, false);
        }
    }

    // D layout: VGPR r holds M=r (lanes 0-15) / M=r+8 (lanes 16-31), N=lane&15
    const int half = lane >> 4;
#pragma unroll
    for (int j = 0; j < NT; ++j) {
#pragma unroll
        for (int r = 0; r < 8; ++r) {
            int mo = tile * 16 + r + 8 * half;
            if (mo < N) H[(size_t)mo * DOUT + j * 16 + nloc] = acc[j][r];
        }
    }
}

// ---------------------------------------------------------------------------
// Aggregation init: Y = selfloop_weight * H + bias
//   smooth self-w: dinv_s[i]^2        sharp self-w: 2 * dinv_h[i]^2
// ---------------------------------------------------------------------------
__global__ void k_agg_init(const float* __restrict__ H, const float* __restrict__ dinv,
                           const float* __restrict__ bias, float* __restrict__ Y,
                           int n, int shift, int sharp) {
    long long i = (long long)blockIdx.x * blockDim.x + threadIdx.x;
    long long tot = (long long)n << shift;
    if (i >= tot) return;
    int node = (int)(i >> shift);
    int f    = (int)(i & ((1 << shift) - 1));
    float dv = dinv[node];
    float sw = dv * dv;
    if (sharp) sw *= 2.0f;
    Y[i] = sw * H[i] + bias[f];
}

// ---------------------------------------------------------------------------
// Edge scatter-add: Y[col] += w(row,col) * H[row]
//   smooth: w =  dinv_s[row]*dinv_s[col]
//   sharp : w = -dinv_h[row]*dinv_h[col]
// tid -> (edge, feature) with feature in low bits: 32 lanes share one edge,
// gathers/atomics are contiguous -> resolved in the 192MB L2.
// ---------------------------------------------------------------------------
__global__ void k_agg_edges(const float* __restrict__ H, const int* __restrict__ row,
                            const int* __restrict__ col, const float* __restrict__ dinv,
                            float* __restrict__ Y, int e, int shift, int sharp) {
    long long i = (long long)blockIdx.x * blockDim.x + threadIdx.x;
    long long tot = (long long)e << shift;
    if (i >= tot) return;
    int eg = (int)(i >> shift);
    int f  = (int)(i & ((1 << shift) - 1));
    int r  = row[eg];
    int c  = col[eg];
    float w = dinv[r] * dinv[c];
    if (sharp) w = -w;
    int dout = 1 << shift;
    float v = w * H[(size_t)r * dout + f];
    unsafeAtomicAdd(&Y[(size_t)c * dout + f], v);   // -> global_atomic_add_f32 in L2
}

// ---------------------------------------------------------------------------
// Host-side orchestration
// ---------------------------------------------------------------------------
static inline int cdiv(long long a, int b) { return (int)((a + b - 1) / b); }

extern "C" void kernel_launch(void* const* d_in, const int* in_sizes, int n_in,
                              void* d_out, int out_size, void* d_ws, size_t ws_size,
                              hipStream_t stream) {
    const float* x   = (const float*)d_in[0];
    const int*   ei  = (const int*)d_in[1];
    const float* We1 = (const float*)d_in[2];  const float* be1 = (const float*)d_in[3];
    const float* We2 = (const float*)d_in[4];  const float* be2 = (const float*)d_in[5];
    const float* We3 = (const float*)d_in[6];  const float* be3 = (const float*)d_in[7];
    const float* Wd1 = (const float*)d_in[8];  const float* bd1 = (const float*)d_in[9];
    const float* Wd2 = (const float*)d_in[10]; const float* bd2 = (const float*)d_in[11];
    const float* Wd3 = (const float*)d_in[12]; const float* bd3 = (const float*)d_in[13];

    const int N = in_sizes[0] / 128;   // x is (N, 128)
    const int E = in_sizes[1] / 2;     // edge_index is (2, E)
    const int* rowp = ei;
    const int* colp = ei + E;

    // workspace carve-up (all 4-byte aligned)
    char* ws = (char*)d_ws;
    int*   indeg  = (int*)ws;    ws += sizeof(int)   * (size_t)N;
    float* dinv_s = (float*)ws;  ws += sizeof(float) * (size_t)N;
    float* dinv_h = (float*)ws;  ws += sizeof(float) * (size_t)N;
    float* hbuf   = (float*)ws;  ws += sizeof(float) * (size_t)N * 128;  // GEMM out (max 128)
    float* ybuf   = (float*)ws;                                          // agg out (max 64)
    float* outp   = (float*)d_out;

    const int TB = 256;
    const int rtiles  = (N + 15) >> 4;
    const int gblocks = (rtiles + 7) >> 3;   // 8 waves (= 8 row-tiles) per 256-thread block

    // ---- normalization prologue ----
    k_zero_i32   <<<cdiv(N, TB), TB, 0, stream>>>(indeg, N);
    k_count_indeg<<<cdiv(E, TB), TB, 0, stream>>>(colp, indeg, E);
    k_dinv       <<<cdiv(N, TB), TB, 0, stream>>>(indeg, dinv_s, dinv_h, N);

    // ---- encoder (smooth norm) ----
    // L1: 128 -> 64
    k_gemm_wmma<128, 4, false><<<gblocks, TB, 0, stream>>>(x, We1, hbuf, N);
    k_agg_init <<<cdiv((long long)N << 6, TB), TB, 0, stream>>>(hbuf, dinv_s, be1, ybuf, N, 6, 0);
    k_agg_edges<<<cdiv((long long)E << 6, TB), TB, 0, stream>>>(hbuf, rowp, colp, dinv_s, ybuf, E, 6, 0);
    // L2: 64 -> 32 (ReLU fused into A-load)
    k_gemm_wmma<64, 2, true><<<gblocks, TB, 0, stream>>>(ybuf, We2, hbuf, N);
    k_agg_init <<<cdiv((long long)N << 5, TB), TB, 0, stream>>>(hbuf, dinv_s, be2, ybuf, N, 5, 0);
    k_agg_edges<<<cdiv((long long)E << 5, TB), TB, 0, stream>>>(hbuf, rowp, colp, dinv_s, ybuf, E, 5, 0);
    // L3: 32 -> 16
    k_gemm_wmma<32, 1, true><<<gblocks, TB, 0, stream>>>(ybuf, We3, hbuf, N);
    k_agg_init <<<cdiv((long long)N << 4, TB), TB, 0, stream>>>(hbuf, dinv_s, be3, ybuf, N, 4, 0);
    k_agg_edges<<<cdiv((long long)E << 4, TB), TB, 0, stream>>>(hbuf, rowp, colp, dinv_s, ybuf, E, 4, 0);

    // ---- decoder (sharp norm) ----
    // L4: 16 -> 32
    k_gemm_wmma<16, 2, true><<<gblocks, TB, 0, stream>>>(ybuf, Wd1, hbuf, N);
    k_agg_init <<<cdiv((long long)N << 5, TB), TB, 0, stream>>>(hbuf, dinv_h, bd1, ybuf, N, 5, 1);
    k_agg_edges<<<cdiv((long long)E << 5, TB), TB, 0, stream>>>(hbuf, rowp, colp, dinv_h, ybuf, E, 5, 1);
    // L5: 32 -> 64
    k_gemm_wmma<32, 4, true><<<gblocks, TB, 0, stream>>>(ybuf, Wd2, hbuf, N);
    k_agg_init <<<cdiv((long long)N << 6, TB), TB, 0, stream>>>(hbuf, dinv_h, bd2, ybuf, N, 6, 1);
    k_agg_edges<<<cdiv((long long)E << 6, TB), TB, 0, stream>>>(hbuf, rowp, colp, dinv_h, ybuf, E, 6, 1);
    // L6: 64 -> 128, no final ReLU, straight to d_out
    k_gemm_wmma<64, 8, true><<<gblocks, TB, 0, stream>>>(ybuf, Wd3, hbuf, N);
    k_agg_init <<<cdiv((long long)N << 7, TB), TB, 0, stream>>>(hbuf, dinv_h, bd3, outp, N, 7, 1);
    k_agg_edges<<<cdiv((long long)E << 7, TB), TB, 0, stream>>>(hbuf, rowp, colp, dinv_h, outp, E, 7, 1);
}